// GCNClassifier_78237124263938
// MI455X (gfx1250) — compile-verified
//
#include <hip/hip_runtime.h>

#define IN_FEAT   67
#define IN_PAD    68      // agg1 padded so k/k+1 loads at k=66 are safe (col 67 zeroed)
#define HIDDEN    256
#define N_CLASS   18
#define NEG_SLOPE 0.01f
#define MT_PER_BLOCK 8    // 8 waves per block, one M-tile each, shared N-tile weights in LDS

typedef float v2f __attribute__((ext_vector_type(2)));
typedef float v8f __attribute__((ext_vector_type(8)));

// ---------------------------------------------------------------------------
// Degree: deg[dst[e]] += 1
// ---------------------------------------------------------------------------
__global__ void deg_kernel(const int* __restrict__ dst, float* __restrict__ deg, int nEdges) {
    int e = blockIdx.x * blockDim.x + threadIdx.x;
    if (e < nEdges) atomicAdd(&deg[dst[e]], 1.0f);
}

// ---------------------------------------------------------------------------
// Neighbor sum: agg[dst[e], :] += feat[src[e], :]  (wave per edge, lane-strided)
// ---------------------------------------------------------------------------
__global__ void scatter_add_kernel(const float* __restrict__ feat,
                                   const int* __restrict__ src,
                                   const int* __restrict__ dst,
                                   float* __restrict__ agg,
                                   int nEdges, int F, int inStride, int outStride) {
    int wave = blockIdx.x * (blockDim.x >> 5) + (threadIdx.x >> 5);
    int lane = threadIdx.x & 31;
    if (wave >= nEdges) return;
    long s = src[wave];
    long d = dst[wave];
    const float* fin  = feat + s * inStride;
    float*       fout = agg  + d * outStride;
    for (int k = lane; k < F; k += 32) atomicAdd(&fout[k], fin[k]);
}

// ---------------------------------------------------------------------------
// gcnt[graph_ids[i]] += 1
// ---------------------------------------------------------------------------
__global__ void gcnt_kernel(const int* __restrict__ gid, float* __restrict__ gcnt, int nNodes) {
    int i = blockIdx.x * blockDim.x + threadIdx.x;
    if (i < nNodes) atomicAdd(&gcnt[gid[i]], 1.0f);
}

// ---------------------------------------------------------------------------
// Layer 1: x1 = leaky( h @ Wself + (agg1/deg) @ Wneigh + b )
// Block = 8 waves = 8 M-tiles sharing one N-tile. Weight strips (K x 16, both
// matrices) staged once into LDS; waves issue V_WMMA_F32_16X16X4_F32 with
// A-frags from VMEM (b64-combinable) and B-frags from LDS.
// A-frag (16x4 f32): lane L: m=L%16, k0 = kb + (L>=16 ? 2 : 0)
// C/D: lane L holds col n=L%16, rows mbase+(0..7), mbase = (L>=16)?8:0.
// ---------------------------------------------------------------------------
__global__ void gemm1_kernel(const float* __restrict__ h,
                             const float* __restrict__ agg1,   // [N, IN_PAD], col 67 = 0
                             const float* __restrict__ deg,
                             const float* __restrict__ Wself,  // [67,256]
                             const float* __restrict__ Wneigh, // [67,256]
                             const float* __restrict__ bias,   // [256]
                             float* __restrict__ x1,           // [N,256]
                             int nNodes, int mTiles) {
    __shared__ float sWs[IN_PAD * 16];
    __shared__ float sWn[IN_PAD * 16];

    int nt      = blockIdx.x & 15;            // 16 N-tiles
    int mtBase  = (blockIdx.x >> 4) * MT_PER_BLOCK;
    int colBase = nt * 16;

    // Stage weight strips (zero-padded to K=68) into LDS.
    for (int i = threadIdx.x; i < IN_PAD * 16; i += blockDim.x) {
        int k = i >> 4, c = i & 15;
        float ws = 0.0f, wn = 0.0f;
        if (k < IN_FEAT) {
            ws = Wself [(long)k * HIDDEN + colBase + c];
            wn = Wneigh[(long)k * HIDDEN + colBase + c];
        }
        sWs[i] = ws;
        sWn[i] = wn;
    }
    __syncthreads();

    int mt = mtBase + (threadIdx.x >> 5);
    if (mt >= mTiles) return;                 // wave-uniform; EXEC stays all-ones

    int lane = threadIdx.x & 31;
    int half = lane >> 4;                     // 0 | 1
    int l16  = lane & 15;
    int row  = mt * 16 + l16;
    int rowc = row < nNodes ? row : nNodes - 1;
    int col  = colBase + l16;

    float rdeg = __builtin_amdgcn_rcpf(fmaxf(deg[rowc], 1.0f));  // exact-int deg -> v_rcp_f32
    const float* hrow = h    + (long)rowc * IN_FEAT;
    const float* arow = agg1 + (long)rowc * IN_PAD;

    v8f acc = {};
    for (int kb = 0; kb < IN_PAD; kb += 4) {
        int k0 = kb + half * 2;
        v2f aS, aN, bS, bN;
        aS.x = (k0     < IN_FEAT) ? hrow[k0]     : 0.0f;
        aS.y = (k0 + 1 < IN_FEAT) ? hrow[k0 + 1] : 0.0f;
        aN.x = arow[k0]     * rdeg;           // padded; col 67 is zero
        aN.y = arow[k0 + 1] * rdeg;
        bS.x = sWs[ k0      * 16 + l16];
        bS.y = sWs[(k0 + 1) * 16 + l16];
        bN.x = sWn[ k0      * 16 + l16];
        bN.y = sWn[(k0 + 1) * 16 + l16];
        acc = __builtin_amdgcn_wmma_f32_16x16x4_f32(false, aS, false, bS, (short)0, acc, false, false);
        acc = __builtin_amdgcn_wmma_f32_16x16x4_f32(false, aN, false, bN, (short)0, acc, false, false);
    }

    float b = bias[col];
    #pragma unroll
    for (int r = 0; r < 8; ++r) {
        int m = mt * 16 + half * 8 + r;
        if (m < nNodes) {
            float v = acc[r] + b;
            v = (v > 0.0f) ? v : NEG_SLOPE * v;
            x1[(long)m * HIDDEN + col] = v;
        }
    }
}

// ---------------------------------------------------------------------------
// Layer 2 fused with graph pooling:
//   val = leaky( x1 @ W2s + (agg2/deg) @ W2n + b2 );  gsum[gid[m], n] += val
// Same LDS weight staging (2 x 256 x 16 f32 = 32KB). The [N,256] layer-2
// activation is never materialized (saves ~200MB of HBM traffic).
// ---------------------------------------------------------------------------
__global__ void gemm2_pool_kernel(const float* __restrict__ x1,    // [N,256]
                                  const float* __restrict__ agg2,  // [N,256]
                                  const float* __restrict__ deg,
                                  const float* __restrict__ Wself, // [256,256]
                                  const float* __restrict__ Wneigh,// [256,256]
                                  const float* __restrict__ bias,  // [256]
                                  const int*  __restrict__ gid,    // [N]
                                  float* __restrict__ gsum,        // [G,256]
                                  int nNodes, int mTiles) {
    __shared__ float sWs[HIDDEN * 16];        // 16 KB
    __shared__ float sWn[HIDDEN * 16];        // 16 KB

    int nt      = blockIdx.x & 15;
    int mtBase  = (blockIdx.x >> 4) * MT_PER_BLOCK;
    int colBase = nt * 16;

    for (int i = threadIdx.x; i < HIDDEN * 16; i += blockDim.x) {
        int k = i >> 4, c = i & 15;
        sWs[i] = Wself [(long)k * HIDDEN + colBase + c];
        sWn[i] = Wneigh[(long)k * HIDDEN + colBase + c];
    }
    __syncthreads();

    int mt = mtBase + (threadIdx.x >> 5);
    if (mt >= mTiles) return;

    int lane = threadIdx.x & 31;
    int half = lane >> 4;
    int l16  = lane & 15;
    int row  = mt * 16 + l16;
    int rowc = row < nNodes ? row : nNodes - 1;
    int col  = colBase + l16;

    float rdeg = __builtin_amdgcn_rcpf(fmaxf(deg[rowc], 1.0f));
    const float* xrow = x1   + (long)rowc * HIDDEN;
    const float* arow = agg2 + (long)rowc * HIDDEN;

    v8f acc = {};
    for (int kb = 0; kb < HIDDEN; kb += 4) {
        int k0 = kb + half * 2;
        v2f aS, aN, bS, bN;
        aS.x = xrow[k0];
        aS.y = xrow[k0 + 1];
        aN.x = arow[k0]     * rdeg;
        aN.y = arow[k0 + 1] * rdeg;
        bS.x = sWs[ k0      * 16 + l16];
        bS.y = sWs[(k0 + 1) * 16 + l16];
        bN.x = sWn[ k0      * 16 + l16];
        bN.y = sWn[(k0 + 1) * 16 + l16];
        acc = __builtin_amdgcn_wmma_f32_16x16x4_f32(false, aS, false, bS, (short)0, acc, false, false);
        acc = __builtin_amdgcn_wmma_f32_16x16x4_f32(false, aN, false, bN, (short)0, acc, false, false);
    }

    float b = bias[col];
    #pragma unroll
    for (int r = 0; r < 8; ++r) {
        int m = mt * 16 + half * 8 + r;
        if (m < nNodes) {
            float v = acc[r] + b;
            v = (v > 0.0f) ? v : NEG_SLOPE * v;
            atomicAdd(&gsum[(long)gid[m] * HIDDEN + col], v);   // L2-resident 256KB
        }
    }
}

// ---------------------------------------------------------------------------
// Final classifier: out[g,c] = (gsum[g,:]/max(gcnt,1)) @ Wfc + bfc   (tiny)
// ---------------------------------------------------------------------------
__global__ void fc_kernel(const float* __restrict__ gsum,
                          const float* __restrict__ gcnt,
                          const float* __restrict__ Wfc,   // [256,18]
                          const float* __restrict__ bfc,   // [18]
                          float* __restrict__ out,         // [G,18]
                          int nGraphs) {
    int g = blockIdx.x;
    int c = threadIdx.x;
    if (g >= nGraphs || c >= N_CLASS) return;
    float inv = 1.0f / fmaxf(gcnt[g], 1.0f);
    float s = 0.0f;
    for (int k = 0; k < HIDDEN; ++k)
        s += gsum[(long)g * HIDDEN + k] * Wfc[(long)k * N_CLASS + c];
    out[(long)g * N_CLASS + c] = s * inv + bfc[c];
}

// ---------------------------------------------------------------------------
extern "C" void kernel_launch(void* const* d_in, const int* in_sizes, int n_in,
                              void* d_out, int out_size, void* d_ws, size_t ws_size,
                              hipStream_t stream) {
    const float* h      = (const float*)d_in[0];
    const float* W1s    = (const float*)d_in[1];
    const float* W1n    = (const float*)d_in[2];
    const float* b1     = (const float*)d_in[3];
    const float* W2s    = (const float*)d_in[4];
    const float* W2n    = (const float*)d_in[5];
    const float* b2     = (const float*)d_in[6];
    const float* Wfc    = (const float*)d_in[7];
    const float* bfc    = (const float*)d_in[8];
    const int*   src    = (const int*)d_in[9];
    const int*   dst    = (const int*)d_in[10];
    const int*   gid    = (const int*)d_in[11];

    const int nNodes  = in_sizes[0] / IN_FEAT;     // 100000
    const int nEdges  = in_sizes[9];               // 1600000
    const int nGraphs = out_size / N_CLASS;        // 256
    const int mTiles  = (nNodes + 15) / 16;

    // Workspace layout (floats): ~206 MB total
    float* ws     = (float*)d_ws;
    float* deg    = ws;                                    // [N]
    float* x1     = deg  + (size_t)nNodes;                 // [N,256]
    float* aggbuf = x1   + (size_t)nNodes * HIDDEN;        // [N,256] (layer1 uses [N,68])
    float* gsum   = aggbuf + (size_t)nNodes * HIDDEN;      // [G,256]
    float* gcnt   = gsum + (size_t)nGraphs * HIDDEN;       // [G]
    (void)ws_size;

    // ---- zero init (graph-capturable memsets) ----
    hipMemsetAsync(deg,  0, (size_t)nNodes * sizeof(float), stream);
    hipMemsetAsync(aggbuf, 0, (size_t)nNodes * IN_PAD * sizeof(float), stream);
    hipMemsetAsync(gsum, 0, (size_t)nGraphs * HIDDEN * sizeof(float), stream);
    hipMemsetAsync(gcnt, 0, (size_t)nGraphs * sizeof(float), stream);

    // ---- degree + graph counts ----
    deg_kernel<<<(nEdges + 255) / 256, 256, 0, stream>>>(dst, deg, nEdges);
    gcnt_kernel<<<(nNodes + 255) / 256, 256, 0, stream>>>(gid, gcnt, nNodes);

    // ---- layer 1 aggregate + GEMM ----
    const int wavesPerBlock = 8; // 256 threads
    int gatherBlocks  = (nEdges + wavesPerBlock - 1) / wavesPerBlock;
    scatter_add_kernel<<<gatherBlocks, 256, 0, stream>>>(h, src, dst, aggbuf,
                                                         nEdges, IN_FEAT, IN_FEAT, IN_PAD);

    int mtBlocks   = (mTiles + MT_PER_BLOCK - 1) / MT_PER_BLOCK;
    int gemmBlocks = mtBlocks * (HIDDEN / 16);
    gemm1_kernel<<<gemmBlocks, 256, 0, stream>>>(h, aggbuf, deg, W1s, W1n, b1, x1,
                                                 nNodes, mTiles);

    // ---- layer 2 aggregate (reuse aggbuf) + fused GEMM/pool ----
    hipMemsetAsync(aggbuf, 0, (size_t)nNodes * HIDDEN * sizeof(float), stream);
    scatter_add_kernel<<<gatherBlocks, 256, 0, stream>>>(x1, src, dst, aggbuf,
                                                         nEdges, HIDDEN, HIDDEN, HIDDEN);
    gemm2_pool_kernel<<<gemmBlocks, 256, 0, stream>>>(x1, aggbuf, deg, W2s, W2n, b2,
                                                      gid, gsum, nNodes, mTiles);

    // ---- classifier ----
    fc_kernel<<<nGraphs, 32, 0, stream>>>(gsum, gcnt, Wfc, bfc, (float*)d_out, nGraphs);
}